// TverskyLayer_58626303590625
// MI455X (gfx1250) — compile-verified
//
#include <hip/hip_runtime.h>

// ---------------------------------------------------------------------------
// TverskyLayer for MI455X (gfx1250, wave32, WMMA)
//   B=256 (batch), P=256 (prototypes), F=512 (features), D=1024 (dim)
// Phase 1: AF = x @ W^T, PF = prototypes @ W^T  (fp32 WMMA 16x16x4),
//          fused epilogue computes ind() and ind()-weighted copies.
// Phase 2: common = AW @ PW^T                   (fp32 WMMA 16x16x4)
// Phase 3: pairwise indicator reduction, 32x32 tile / 2x2 micro-tile,
//          float2-interleaved LDS, v_tanh_f32-based indicator.
// ---------------------------------------------------------------------------

typedef __attribute__((ext_vector_type(2))) float v2f;
typedef __attribute__((ext_vector_type(8))) float v8f;

#define LOG2E 1.4426950408889634f

// indicator(v) = (tanh(10*(v-0.5))+1)/2
__device__ __forceinline__ float ind_sig(float v) {
#if __has_builtin(__builtin_amdgcn_tanhf)
    return fmaf(0.5f, __builtin_amdgcn_tanhf(fmaf(10.0f, v, -5.0f)), 0.5f);
#else
    // sigmoid(20v-10) = 1/(1+exp(10-20v))
    float t = (10.0f - 20.0f * v) * LOG2E;
    float e = __builtin_amdgcn_exp2f(t);
    return __builtin_amdgcn_rcpf(1.0f + e);
#endif
}

// ip = indicator(d), in = indicator(-d)   (tanh is odd: shares the fma form)
__device__ __forceinline__ void ind_pair(float d, float& ip, float& in) {
#if __has_builtin(__builtin_amdgcn_tanhf)
    const float t1 = __builtin_amdgcn_tanhf(fmaf(10.0f, d, -5.0f));
    const float t2 = __builtin_amdgcn_tanhf(fmaf(10.0f, d,  5.0f));
    ip = fmaf( 0.5f, t1, 0.5f);
    in = fmaf(-0.5f, t2, 0.5f);
#else
    ip = ind_sig(d);
    in = ind_sig(-d);
#endif
}

// ---------------------------------------------------------------------------
// Phase 1: OF[m,n] = sum_k X[m,k]*W[n,k];  OI = ind(OF);  OW = OF*OI
// One wave per 16x16 output tile (ISA 7.12.2 fragment layouts).
// ---------------------------------------------------------------------------
__global__ __launch_bounds__(256) void tversky_feat_gemm(
    const float* __restrict__ X, const float* __restrict__ W,
    float* __restrict__ OF, float* __restrict__ OI, float* __restrict__ OW,
    int M, int N, int K) {
    const int wave = (blockIdx.x * blockDim.x + threadIdx.x) >> 5;
    const int lane = threadIdx.x & 31;
    const int ntile = N >> 4;
    const int tm = wave / ntile;
    const int tn = wave - tm * ntile;
    if (tm * 16 >= M) return;                      // wave-uniform guard

    const int half = lane >> 4;                    // K-pair select
    const int l16  = lane & 15;
    const int mrow = tm * 16 + l16;
    const int ncol = tn * 16 + l16;
    const int koff = half * 2;

    v8f c = {};
    const float* __restrict__ xrow = X + (size_t)mrow * K + koff;
    const float* __restrict__ wrow = W + (size_t)ncol * K + koff;
#pragma unroll 4
    for (int k = 0; k < K; k += 4) {
        v2f a = *(const v2f*)(xrow + k);
        v2f b = *(const v2f*)(wrow + k);
        c = __builtin_amdgcn_wmma_f32_16x16x4_f32(
                false, a, false, b, (short)0, c, false, false);
    }

#pragma unroll
    for (int r = 0; r < 8; ++r) {
        const int m = tm * 16 + r + half * 8;
        const int n = tn * 16 + l16;
        const size_t idx = (size_t)m * N + n;
        const float v = c[r];
        const float iv = ind_sig(v);
        OF[idx] = v;
        OI[idx] = iv;
        OW[idx] = v * iv;
    }
}

// ---------------------------------------------------------------------------
// Phase 2: C[m,n] = sum_k AW[m,k]*PW[n,k]   (M=256, N=256, K=512)
// ---------------------------------------------------------------------------
__global__ __launch_bounds__(256) void tversky_common_gemm(
    const float* __restrict__ AW, const float* __restrict__ PW,
    float* __restrict__ C, int M, int N, int K) {
    const int wave = (blockIdx.x * blockDim.x + threadIdx.x) >> 5;
    const int lane = threadIdx.x & 31;
    const int ntile = N >> 4;
    const int tm = wave / ntile;
    const int tn = wave - tm * ntile;
    if (tm * 16 >= M) return;

    const int half = lane >> 4;
    const int l16  = lane & 15;
    const int mrow = tm * 16 + l16;
    const int ncol = tn * 16 + l16;
    const int koff = half * 2;

    v8f c = {};
    const float* __restrict__ arow = AW + (size_t)mrow * K + koff;
    const float* __restrict__ prow = PW + (size_t)ncol * K + koff;
#pragma unroll 4
    for (int k = 0; k < K; k += 4) {
        v2f a = *(const v2f*)(arow + k);
        v2f b = *(const v2f*)(prow + k);
        c = __builtin_amdgcn_wmma_f32_16x16x4_f32(
                false, a, false, b, (short)0, c, false, false);
    }
#pragma unroll
    for (int r = 0; r < 8; ++r) {
        const int m = tm * 16 + r + half * 8;
        const int n = tn * 16 + l16;
        C[(size_t)m * N + n] = c[r];
    }
}

// ---------------------------------------------------------------------------
// Phase 3: 32x32 (b,p) tile per 256-thread block, 2x2 micro-tile per thread.
// LDS holds (feature, indicator) interleaved as float2 -> ds_load_b64 reads,
// 4 b64 LDS reads feed 4 output accumulations (4x less DS traffic than 1:1).
// ---------------------------------------------------------------------------
#define FCHUNK 32
__global__ __launch_bounds__(256) void tversky_pairwise(
    const float* __restrict__ AF, const float* __restrict__ AI,
    const float* __restrict__ PF, const float* __restrict__ PI,
    const float* __restrict__ COM,
    const float* __restrict__ alpha, const float* __restrict__ beta,
    const float* __restrict__ theta,
    float* __restrict__ out, int Pdim, int F) {
    __shared__ float2 sA[32][FCHUNK + 1];   // rows: b in tile, .x=feat .y=ind
    __shared__ float2 sP[32][FCHUNK + 1];   // rows: p in tile

    const int tid = threadIdx.x;
    const int tx  = tid & 15;               // p micro-tile column
    const int ty  = tid >> 4;               // b micro-tile row
    const int b0  = blockIdx.y * 32;
    const int p0  = blockIdx.x * 32;

    float ad00 = 0.f, ad01 = 0.f, ad10 = 0.f, ad11 = 0.f;
    float bd00 = 0.f, bd01 = 0.f, bd10 = 0.f, bd11 = 0.f;

    for (int f0 = 0; f0 < F; f0 += FCHUNK) {
#pragma unroll
        for (int i = tid; i < 32 * FCHUNK; i += 256) {   // coalesced stage-in
            const int r = i >> 5, cix = i & (FCHUNK - 1);
            const size_t ai_idx = (size_t)(b0 + r) * F + f0 + cix;
            const size_t pi_idx = (size_t)(p0 + r) * F + f0 + cix;
            sA[r][cix] = make_float2(AF[ai_idx], AI[ai_idx]);
            sP[r][cix] = make_float2(PF[pi_idx], PI[pi_idx]);
        }
        __syncthreads();
#pragma unroll 4
        for (int cix = 0; cix < FCHUNK; ++cix) {
            const float2 a0 = sA[2 * ty + 0][cix];
            const float2 a1 = sA[2 * ty + 1][cix];
            const float2 q0 = sP[2 * tx + 0][cix];
            const float2 q1 = sP[2 * tx + 1][cix];

            float d, w, ip, in;
            d = a0.x - q0.x; w = d * (a0.y * q0.y); ind_pair(d, ip, in);
            ad00 = fmaf(w, ip, ad00); bd00 = fmaf(-w, in, bd00);
            d = a0.x - q1.x; w = d * (a0.y * q1.y); ind_pair(d, ip, in);
            ad01 = fmaf(w, ip, ad01); bd01 = fmaf(-w, in, bd01);
            d = a1.x - q0.x; w = d * (a1.y * q0.y); ind_pair(d, ip, in);
            ad10 = fmaf(w, ip, ad10); bd10 = fmaf(-w, in, bd10);
            d = a1.x - q1.x; w = d * (a1.y * q1.y); ind_pair(d, ip, in);
            ad11 = fmaf(w, ip, ad11); bd11 = fmaf(-w, in, bd11);
        }
        __syncthreads();
    }

    const float th = theta[0], al = alpha[0], be = beta[0];
    const int b_0 = b0 + 2 * ty, p_0 = p0 + 2 * tx;
    {
        size_t i00 = (size_t)(b_0 + 0) * Pdim + (p_0 + 0);
        size_t i01 = (size_t)(b_0 + 0) * Pdim + (p_0 + 1);
        size_t i10 = (size_t)(b_0 + 1) * Pdim + (p_0 + 0);
        size_t i11 = (size_t)(b_0 + 1) * Pdim + (p_0 + 1);
        out[i00] = th * COM[i00] - al * ad00 - be * bd00;
        out[i01] = th * COM[i01] - al * ad01 - be * bd01;
        out[i10] = th * COM[i10] - al * ad10 - be * bd10;
        out[i11] = th * COM[i11] - al * ad11 - be * bd11;
    }
}

// ---------------------------------------------------------------------------
extern "C" void kernel_launch(void* const* d_in, const int* in_sizes, int n_in,
                              void* d_out, int out_size, void* d_ws, size_t ws_size,
                              hipStream_t stream) {
    (void)in_sizes; (void)n_in; (void)out_size; (void)ws_size;
    const int B = 256, P = 256, F = 512, D = 1024;

    const float* x          = (const float*)d_in[0];   // [B, D]
    const float* features   = (const float*)d_in[1];   // [F, D]
    const float* prototypes = (const float*)d_in[2];   // [P, D]
    const float* alpha      = (const float*)d_in[3];
    const float* beta       = (const float*)d_in[4];
    const float* theta      = (const float*)d_in[5];
    float* out = (float*)d_out;                        // [B, P]

    float* ws = (float*)d_ws;
    const size_t BF = (size_t)B * F;                   // 131072
    float* AF = ws;            // a_features            [B,F]
    float* AI = AF + BF;       // ind(a_features)       [B,F]
    float* AW = AI + BF;       // a_features*ind        [B,F]
    float* PF = AW + BF;       // p_features            [P,F]
    float* PI = PF + BF;       // ind(p_features)       [P,F]
    float* PW = PI + BF;       // p_features*ind        [P,F]
    float* COM = PW + BF;      // common                [B,P]

    // Phase 1: two feature GEMMs (one wave / 16x16 tile, 8 waves / block)
    {
        const int tiles = (B / 16) * (F / 16);         // 512
        tversky_feat_gemm<<<tiles / 8, 256, 0, stream>>>(x, features, AF, AI, AW, B, F, D);
        tversky_feat_gemm<<<tiles / 8, 256, 0, stream>>>(prototypes, features, PF, PI, PW, P, F, D);
    }
    // Phase 2: common GEMM
    {
        const int tiles = (B / 16) * (P / 16);         // 256
        tversky_common_gemm<<<tiles / 8, 256, 0, stream>>>(AW, PW, COM, B, P, F);
    }
    // Phase 3: pairwise indicator reduction + fused combine
    {
        dim3 grid(P / 32, B / 32);                     // 8 x 8 = 64 blocks
        dim3 block(256);
        tversky_pairwise<<<grid, block, 0, stream>>>(AF, AI, PF, PI, COM,
                                                     alpha, beta, theta, out, P, F);
    }
}